// PassiveWaveDigitalMixer_22978075034043
// MI455X (gfx1250) — compile-verified
//
#include <hip/hip_runtime.h>
#include <math.h>

// Problem constants (B,C,H,W,O,M) = (4,128,256,256,16,32)
#define HW   65536
#define WD   256
#define CCH  128
#define NB   4

typedef float v2f __attribute__((ext_vector_type(2)));
typedef float v8f __attribute__((ext_vector_type(8)));
typedef float v4f __attribute__((ext_vector_type(4)));
typedef int   v4i_gnu __attribute__((vector_size(16)));   // matches builtin's V4i
typedef __attribute__((address_space(1))) v4i_gnu* g_v4i_ptr;
typedef __attribute__((address_space(3))) v4i_gnu* l_v4i_ptr;

#if __has_builtin(__builtin_amdgcn_global_load_async_to_lds_b128) && \
    __has_builtin(__builtin_amdgcn_s_wait_asynccnt)
#define PWDM_ASYNC_LDS 1
#endif

__device__ __forceinline__ float softplus_eps(float v) {
    // stable softplus + EPS
    return fmaxf(v, 0.0f) + log1pf(expf(-fabsf(v))) + 1e-4f;
}
__device__ __forceinline__ float clamp01(float v) {
    // GAIN_CAP - relu(GAIN_CAP - relu(v)) == clamp to [0, 0.99]
    return fminf(fmaxf(v, 0.0f), 0.99f);
}

// ---------------------------------------------------------------------------
// Kernel 1: per-channel / per-(b,c) parameter precompute into workspace.
// ws layout (floats): [0..511] ghm  [512..1023] gvm  [1024..1535] gdm
//                     [1536..1663] rih  [1664..1791] riv  [1792..1919] rid
//                     [2048 ...] steer fac fields fac[b][p][HW]  (p = 0..2)
// ---------------------------------------------------------------------------
__global__ __launch_bounds__(128) void pwdm_params(
    const float* __restrict__ modulation, const float* __restrict__ mod_w,
    const float* __restrict__ r_c, const float* __restrict__ r_h,
    const float* __restrict__ r_v, const float* __restrict__ r_d,
    const float* __restrict__ g_h, const float* __restrict__ g_v,
    const float* __restrict__ g_d, float* __restrict__ ws)
{
    int c = threadIdx.x;  // 128 threads, one per channel
    float pc = softplus_eps(r_c[c]);
    ws[1536 + c] = 1.0f / (pc + softplus_eps(r_h[c]));
    ws[1664 + c] = 1.0f / (pc + softplus_eps(r_v[c]));
    ws[1792 + c] = 1.0f / (2.0f * (pc + softplus_eps(r_d[c])));
    float th = tanhf(g_h[c]);
    float tv = tanhf(g_v[c]);
    float td = tanhf(g_d[c]);
    for (int b = 0; b < NB; ++b) {
        float d0 = 0.0f, d1 = 0.0f, d2 = 0.0f;
        #pragma unroll
        for (int m = 0; m < 32; ++m) {
            float mv = modulation[b * 32 + m];
            d0 = fmaf(mv, mod_w[ 0 + m], d0);
            d1 = fmaf(mv, mod_w[32 + m], d1);
            d2 = fmaf(mv, mod_w[64 + m], d2);
        }
        d0 = tanhf(d0); d1 = tanhf(d1); d2 = tanhf(d2);
        ws[       b * CCH + c] = clamp01(th * (1.0f + 0.1f * d0));
        ws[ 512 + b * CCH + c] = clamp01(tv * (1.0f + 0.1f * d1));
        ws[1024 + b * CCH + c] = clamp01(td * (1.0f + 0.1f * d2));
    }
}

// ---------------------------------------------------------------------------
// Kernel 2: steer 1x1 conv + tanh via V_WMMA_F32_16X16X4_F32 (exact f32 path).
// One wave = 16 pixels. D(16x16) = A(16 pixels x 4 chan) * B(4 chan x 16 out),
// accumulated over 4 K-chunks to cover O=16 input channels. Only N=0..2 used.
// fac[b][p][pix] = 1 + 0.2*tanh(sum_o steer_w[p][o]*steer[b][o][pix])
// B-weight loads use a wrapped (n&3) index so they are always in-bounds and
// branchless; the N>=4 zero padding becomes a v_cndmask select, not exec flow.
// ---------------------------------------------------------------------------
__global__ __launch_bounds__(256) void pwdm_steer(
    const float* __restrict__ steer, const float* __restrict__ steer_w,
    float* __restrict__ fac)
{
    int wave    = (blockIdx.x * blockDim.x + threadIdx.x) >> 5;
    int lane    = threadIdx.x & 31;
    int b       = wave >> 12;             // 4096 pixel-tiles per batch
    int pixbase = (wave & 4095) << 4;
    int n       = lane & 15;              // A: M index; B/D: N index
    int half    = lane >> 4;              // lanes 16-31 carry K+2 / M+8
    int koff    = half * 2;
    bool valid  = (n < 4);
    const float* wrow = steer_w + (n & 3) * 16;  // always in-bounds

    const float* sp = steer + (size_t)b * 16 * HW + pixbase + n;
    v8f acc = {};
    #pragma unroll
    for (int k0 = 0; k0 < 16; k0 += 4) {
        int ch = k0 + koff;
        v2f a, bm;
        // A (16x4 f32): lane m holds (K=ch, K=ch+1); lanes>=16 hold K+2
        a.x = sp[(size_t)ch * HW];
        a.y = sp[(size_t)(ch + 1) * HW];
        // B (4x16 f32): lane n holds w^T[K=ch..ch+1][n]; pad N>=4 with 0
        float w0 = wrow[ch];
        float w1 = wrow[ch + 1];
        bm.x = valid ? w0 : 0.0f;
        bm.y = valid ? w1 : 0.0f;
        acc = __builtin_amdgcn_wmma_f32_16x16x4_f32(
            false, a, false, bm, (short)0, acc, false, false);
    }
    // D layout: vgpr j = pixel (j + 8*half), lane%16 = output field p
    if (n < 3) {
        float* op = fac + ((size_t)b * 3 + n) * HW + pixbase + half * 8;
        float4 lo, hi;
        lo.x = 1.0f + 0.2f * tanhf(acc[0]);
        lo.y = 1.0f + 0.2f * tanhf(acc[1]);
        lo.z = 1.0f + 0.2f * tanhf(acc[2]);
        lo.w = 1.0f + 0.2f * tanhf(acc[3]);
        hi.x = 1.0f + 0.2f * tanhf(acc[4]);
        hi.y = 1.0f + 0.2f * tanhf(acc[5]);
        hi.z = 1.0f + 0.2f * tanhf(acc[6]);
        hi.w = 1.0f + 0.2f * tanhf(acc[7]);
        *(float4*)op       = lo;
        *(float4*)(op + 4) = hi;
    }
}

// ---------------------------------------------------------------------------
// Kernel 3: memory-bound periodic 3x3 star stencil.
// One block per (plane, 16-row band). 18 halo rows staged to LDS via
// GLOBAL_LOAD_ASYNC_TO_LDS_B128 (ASYNCcnt path; x read exactly once from HBM,
// no VGPR round-trip), wrap handled by &255 index masks. Output uses
// non-temporal stores so the write-once stream doesn't evict the hot
// steer-fac arrays from L2.
// ---------------------------------------------------------------------------
__device__ __forceinline__ float pixel_update(
    float c, float l, float r, float u, float d,
    float ulv, float urv, float dlv, float drv,
    float fH, float fV, float fD,
    float ghm, float gvm, float gdm, float rih, float riv, float rid)
{
    float th = (l - c) + (r - c);
    float tv = (u - c) + (d - c);
    float td = ((ulv - c) + (urv - c)) + ((dlv - c) + (drv - c));
    float gh = clamp01(ghm * fH);
    float gv = clamp01(gvm * fV);
    float gd = clamp01(gdm * fD);
    return c + gh * (th * rih) + gv * (tv * riv) + gd * (td * rid);
}

__global__ __launch_bounds__(256) void pwdm_stencil(
    const float* __restrict__ x, const float* __restrict__ ws,
    float* __restrict__ out)
{
    const float* fac = ws + 2048;
    int band  = blockIdx.x & 15;          // 16 bands of 16 rows
    int plane = blockIdx.x >> 4;          // 512 (b,c) planes
    int b = plane >> 7;
    int c = plane & 127;

    const float* xp = x   + (size_t)plane * HW;
    float*       op = out + (size_t)plane * HW;
    const float* fh = fac + ((size_t)b * 3 + 0) * HW;
    const float* fv = fac + ((size_t)b * 3 + 1) * HW;
    const float* fd = fac + ((size_t)b * 3 + 2) * HW;

    float ghm = ws[       b * CCH + c];
    float gvm = ws[ 512 + b * CCH + c];
    float gdm = ws[1024 + b * CCH + c];
    float rih = ws[1536 + c];
    float riv = ws[1664 + c];
    float rid = ws[1792 + c];

    __shared__ __align__(16) float S[18 * WD];
    int tid = threadIdx.x;
    int h0  = band << 4;

    // Warm the fac rows for this band (gfx1250 global_prefetch_b8 path).
    if (tid < 16) {
        __builtin_prefetch(fh + (h0 + tid) * WD, 0, 0);
        __builtin_prefetch(fv + (h0 + tid) * WD, 0, 0);
        __builtin_prefetch(fd + (h0 + tid) * WD, 0, 0);
    }

    // Stage rows h0-1 .. h0+16 (periodic) into LDS, 16B per lane per issue.
#ifdef PWDM_ASYNC_LDS
    for (int i = tid; i < 18 * 64; i += 256) {
        int r = i >> 6, q = i & 63;
        int h = (h0 - 1 + r) & 255;
        __builtin_amdgcn_global_load_async_to_lds_b128(
            (g_v4i_ptr)(xp + h * WD + q * 4),
            (l_v4i_ptr)(S + r * WD + q * 4),
            0, 0);
    }
    __builtin_amdgcn_s_wait_asynccnt(0);   // own LDS writes landed
#else
    for (int i = tid; i < 18 * 64; i += 256) {
        int r = i >> 6, q = i & 63;
        int h = (h0 - 1 + r) & 255;
        *(float4*)(S + r * WD + q * 4) = *(const float4*)(xp + h * WD + q * 4);
    }
#endif
    __syncthreads();

    #pragma unroll
    for (int it = 0; it < 4; ++it) {
        int u  = tid + (it << 8);
        int r  = u >> 6, q = u & 63;
        int w4 = q << 2;
        int h  = h0 + r;
        const float* Rc = S + (r + 1) * WD;
        const float* Ru = S + (r    ) * WD;
        const float* Rd = S + (r + 2) * WD;
        float4 mc = *(const float4*)(Rc + w4);
        float4 mu = *(const float4*)(Ru + w4);
        float4 md = *(const float4*)(Rd + w4);
        int wl = (w4 - 1) & 255, wr = (w4 + 4) & 255;  // periodic in W
        float cl = Rc[wl], cr = Rc[wr];
        float ul = Ru[wl], ur = Ru[wr];
        float dl = Rd[wl], dr = Rd[wr];
        float4 Fh = *(const float4*)(fh + h * WD + w4);
        float4 Fv = *(const float4*)(fv + h * WD + w4);
        float4 Fd = *(const float4*)(fd + h * WD + w4);
        v4f o;   // native vector type: required by __builtin_nontemporal_store
        o.x = pixel_update(mc.x, cl,   mc.y, mu.x, md.x, ul,   mu.y, dl,   md.y,
                           Fh.x, Fv.x, Fd.x, ghm, gvm, gdm, rih, riv, rid);
        o.y = pixel_update(mc.y, mc.x, mc.z, mu.y, md.y, mu.x, mu.z, md.x, md.z,
                           Fh.y, Fv.y, Fd.y, ghm, gvm, gdm, rih, riv, rid);
        o.z = pixel_update(mc.z, mc.y, mc.w, mu.z, md.z, mu.y, mu.w, md.y, md.w,
                           Fh.z, Fv.z, Fd.z, ghm, gvm, gdm, rih, riv, rid);
        o.w = pixel_update(mc.w, mc.z, cr,   mu.w, md.w, mu.z, ur,   md.z, dr,
                           Fh.w, Fv.w, Fd.w, ghm, gvm, gdm, rih, riv, rid);
        __builtin_nontemporal_store(o, (v4f*)(op + h * WD + w4));
    }
}

extern "C" void kernel_launch(void* const* d_in, const int* in_sizes, int n_in,
                              void* d_out, int out_size, void* d_ws, size_t ws_size,
                              hipStream_t stream) {
    const float* x          = (const float*)d_in[0];
    const float* steer      = (const float*)d_in[1];
    const float* modulation = (const float*)d_in[2];
    const float* r_center   = (const float*)d_in[3];
    const float* r_horiz    = (const float*)d_in[4];
    const float* r_vert     = (const float*)d_in[5];
    const float* r_diag     = (const float*)d_in[6];
    const float* g_horiz    = (const float*)d_in[7];
    const float* g_vert     = (const float*)d_in[8];
    const float* g_diag     = (const float*)d_in[9];
    const float* steer_w    = (const float*)d_in[10];
    const float* mod_w      = (const float*)d_in[11];
    float* out = (float*)d_out;
    float* wsf = (float*)d_ws;           // params at [0,1920), fac at +2048 floats
    float* fac = wsf + 2048;

    pwdm_params<<<1, 128, 0, stream>>>(modulation, mod_w, r_center, r_horiz,
                                       r_vert, r_diag, g_horiz, g_vert, g_diag,
                                       wsf);
    // 4 batches * 4096 tiles of 16 pixels = 16384 waves = 2048 blocks * 8 waves
    pwdm_steer<<<2048, 256, 0, stream>>>(steer, steer_w, fac);
    // 512 planes * 16 bands
    pwdm_stencil<<<8192, 256, 0, stream>>>(x, wsf, out);
}